// LearnedEquivarianceFC_75737453297841
// MI455X (gfx1250) — compile-verified
//
#include <hip/hip_runtime.h>
#include <hip/hip_bf16.h>

typedef __attribute__((ext_vector_type(2))) float v2f;
typedef __attribute__((ext_vector_type(8))) float v8f;

#define NGRP 40
#define NB   128
#define DIM  1024
#define MAX_TILES 64

// ws layout (in ints):
//   [0 .. 39]            counts[grp]
//   [64 .. 64+NGRP*NB)   rows[grp*NB + i]  (sample indices per group, ascending)
//   [5184 .. 5184+64)    tile table: (grp<<16)|rowOffset, -1 = unused slot
#define WS_COUNTS 0
#define WS_ROWS   64
#define WS_TILES  5184

// ---------------------------------------------------------------------------
// Prologue: bucket samples by matrix index, build padded-M tile table.
// Fully deterministic (each group scanned in ascending sample order).
// ---------------------------------------------------------------------------
__global__ void build_tiles_kernel(const int* __restrict__ g,
                                   int* __restrict__ ws) {
    __shared__ int sCnt[NGRP];
    const int t = threadIdx.x;
    if (t < NGRP) {
        int cnt = 0;
        for (int b = 0; b < NB; ++b) {
            int grp = g[b] % NGRP;       // g in [0,100), non-negative
            if (grp == t) {
                ws[WS_ROWS + t * NB + cnt] = b;
                ++cnt;
            }
        }
        sCnt[t] = cnt;
        ws[WS_COUNTS + t] = cnt;
    }
    __syncthreads();
    if (t == 0) {
        int nt = 0;
        for (int grp = 0; grp < NGRP; ++grp)
            for (int off = 0; off < sCnt[grp]; off += 16)
                ws[WS_TILES + nt++] = (grp << 16) | off;
        for (; nt < MAX_TILES; ++nt)
            ws[WS_TILES + nt] = -1;
    }
}

// ---------------------------------------------------------------------------
// Grouped GEMM: one block = (one 16-row M-tile of a group) x (128 N columns).
// 8 wave32s; each wave owns one 16-wide N-tile and runs K=1024 in steps of 4
// through V_WMMA_F32_16X16X4_F32. T_g is streamed from HBM exactly once per
// group (memory-bound: ~160 MB total -> ~7 us at 23.3 TB/s).
// ---------------------------------------------------------------------------
__global__ __launch_bounds__(256) void grouped_gemm_kernel(
    const float* __restrict__ x,      // (128, 1024)
    const float* __restrict__ T,      // (40, 1024, 1024)
    const int*   __restrict__ ws,
    float*       __restrict__ out) {  // (128, 1024)

    // A tile staged K-major: Alds[k*16 + r]  (exactly 64 KB)
    __shared__ float Alds[DIM * 16];

    const int tile = ws[WS_TILES + blockIdx.x];
    if (tile < 0) return;                      // block-uniform exit
    const int grp  = tile >> 16;
    const int roff = tile & 0xFFFF;
    int m = ws[WS_COUNTS + grp] - roff;
    if (m > 16) m = 16;

    const int tid = threadIdx.x;

    // Stage gathered, zero-padded A rows into LDS (transposed to K-major).
    // Global reads of x are coalesced along d.
    for (int r = 0; r < 16; ++r) {
        const int rowIdx = (r < m) ? ws[WS_ROWS + grp * NB + roff + r] : -1;
        const float* xr = x + (size_t)(rowIdx < 0 ? 0 : rowIdx) * DIM;
        for (int d = tid; d < DIM; d += 256) {
            float v = (rowIdx >= 0) ? xr[d] : 0.0f;
            Alds[d * 16 + r] = v;
        }
    }
    __syncthreads();

    const int wave = tid >> 5;
    const int lane = tid & 31;
    const int r16  = lane & 15;      // N column within tile / M row for A
    const int half = lane >> 4;      // 0: K+{0,1}, 1: K+{2,3}  (ISA 16x4 f32 A layout)
    const int krow = half * 2;

    const int n0 = (blockIdx.y * 8 + wave) * 16;
    const float* Bp = T + (size_t)grp * DIM * DIM + n0 + r16;

    v8f c = {};

    for (int kk = 0; kk < DIM; kk += 16) {
        // Prefetch B 64 K-rows ahead (speculative; OOB silently dropped).
        __builtin_prefetch(Bp + (size_t)(kk + 64) * DIM, 0, 0);
#pragma unroll
        for (int ks = 0; ks < 16; ks += 4) {
            const int k0 = kk + ks + krow;
            v2f a, b;
            a[0] = Alds[k0 * 16 + r16];           // conflict-free ds_load
            a[1] = Alds[(k0 + 1) * 16 + r16];
            b[0] = Bp[(size_t)k0 * DIM];          // 16 lanes x 4B = one 64B line
            b[1] = Bp[(size_t)(k0 + 1) * DIM];
            c = __builtin_amdgcn_wmma_f32_16x16x4_f32(
                    false, a, false, b, (short)0, c, false, false);
        }
    }

    // D layout: VGPR j holds (M = j + 8*half, N = n0 + r16).
    // Scatter rows back to their gathered sample indices; skip padding rows.
#pragma unroll
    for (int j = 0; j < 8; ++j) {
        const int mrow = j + half * 8;
        if (mrow < m) {
            const int b = ws[WS_ROWS + grp * NB + roff + mrow];
            out[(size_t)b * DIM + n0 + r16] = c[j];
        }
    }
}

// ---------------------------------------------------------------------------
// Host-side launcher
// ---------------------------------------------------------------------------
extern "C" void kernel_launch(void* const* d_in, const int* in_sizes, int n_in,
                              void* d_out, int out_size, void* d_ws, size_t ws_size,
                              hipStream_t stream) {
    const float* x = (const float*)d_in[0];                // (128,1024) f32
    const int*   g = (const int*)d_in[1];                  // (128,) int
    const float* T = (const float*)d_in[2];                // (40,1024,1024) f32
    // d_in[3] = l, d_in[4] = n : unused by the forward pass.
    int*   ws  = (int*)d_ws;
    float* out = (float*)d_out;

    build_tiles_kernel<<<1, 64, 0, stream>>>(g, ws);

    dim3 grid(MAX_TILES, 8);   // 64 tile slots (>= max possible tiles) x 8 N-chunks
    grouped_gemm_kernel<<<grid, 256, 0, stream>>>(x, T, ws, out);
}